// HGT_20091857011148
// MI455X (gfx1250) — compile-verified
//
#include <hip/hip_runtime.h>

#define RA 20000      // author nodes
#define RP 50000      // paper nodes
#define FDIM 128      // HID = IN_CH
#define NHEADS 4
#define EDG 300000
#define OUTC 64

typedef __attribute__((ext_vector_type(2))) float v2f;
typedef __attribute__((ext_vector_type(8))) float v8f;

__device__ inline void atomicMaxF(float* addr, float val) {
    int* ia = (int*)addr;
    int old = __float_as_int(*addr);
    while (__int_as_float(old) < val) {
        int prev = atomicCAS(ia, old, __float_as_int(val));
        if (prev == old) break;
        old = prev;
    }
}

// ------------------------------------------------------------------
// Y[M,N] = epilogue( X[M,128] @ W[128,N] + bias )
// MODE 0: +bias (bias may be null)      MODE 1: relu(+bias)
// MODE 2: gate*( +bias ) + (1-gate)*res   (gate = sigmoid(*gate_p))
// Uses V_WMMA_F32_16X16X4_F32 (exact fp32 matrix pipe).
// A strip preloaded to registers (32 x v2f per lane, all loads in
// flight up front). W staged in LDS pair-swizzled so each B fragment
// is a single ds_load_b64 into an even VGPR pair (no shuffle movs).
// Safe for in-place X==Y: each wave reads only its own 16 rows.
// ------------------------------------------------------------------
template<int N, int MODE>
__global__ __launch_bounds__(256) void gemm128_kernel(
    const float* X, const float* __restrict__ W,
    const float* __restrict__ bias, const float* res,
    const float* __restrict__ gate_p, float* Y, int M)
{
    constexpr int NT = N / 16;
    // pair-swizzled: Wlds[((k>>1)*N + col)*2 + {0,1}] = W[k][col], W[k+1][col]
    __shared__ __align__(16) float Wlds[128 * N];
    const int tid = threadIdx.x;
    for (int i = tid; i < 64 * N; i += 256) {
        const int k2  = i / N;          // N is a power of two -> shift
        const int col = i & (N - 1);
        v2f p;
        p.x = W[(size_t)(2 * k2)     * N + col];
        p.y = W[(size_t)(2 * k2 + 1) * N + col];
        *(v2f*)(&Wlds[(size_t)i * 2]) = p;
    }
    __syncthreads();

    const int lane  = tid & 31;
    const int wave  = tid >> 5;
    const int row0  = blockIdx.x * 128 + wave * 16;
    if (row0 >= M) return;                 // wave-uniform: EXEC stays all-ones
    const int lhalf = lane >> 4;           // 0: K pair {k,k+1}; 1: {k+2,k+3}
    const int l15   = lane & 15;
    const float* xrow = X + (size_t)(row0 + l15) * 128;

    // Preload the lane's entire A strip: {k, k+1} pair of each K-quad.
    v2f afrag[32];
    #pragma unroll
    for (int i = 0; i < 32; i++)
        afrag[i] = *(const v2f*)(xrow + i * 4 + 2 * lhalf);

    v8f acc[NT] = {};
    #pragma unroll
    for (int kk4 = 0; kk4 < 32; kk4++) {
        const int kp = kk4 * 2 + lhalf;    // (ka0 >> 1) where ka0 = kk4*4 + 2*lhalf
        #pragma unroll
        for (int t = 0; t < NT; t++) {
            const int col = t * 16 + l15;
            v2f b = *(const v2f*)(&Wlds[(size_t)(kp * N + col) * 2]);
            acc[t] = __builtin_amdgcn_wmma_f32_16x16x4_f32(
                false, afrag[kk4], false, b, (short)0, acc[t], false, false);
        }
    }

    float g = 1.0f;
    if (MODE == 2) g = 1.0f / (1.0f + expf(-gate_p[0]));
    #pragma unroll
    for (int t = 0; t < NT; t++) {
        const int col = t * 16 + l15;
        const float bv = bias ? bias[col] : 0.0f;
        #pragma unroll
        for (int r = 0; r < 8; r++) {
            const int row = row0 + r + lhalf * 8;    // C/D layout: lanes>=16 -> M+8
            float v = acc[t][r] + bv;
            if (MODE == 1) v = fmaxf(v, 0.0f);
            if (MODE == 2) v = g * v + (1.0f - g) * res[(size_t)row * N + col];
            Y[(size_t)row * N + col] = v;
        }
    }
}

// rel[4][32][32] -> block-diagonal W[128][128]
__global__ void expand_blockdiag_kernel(const float* __restrict__ rel,
                                        float* __restrict__ W)
{
    int i = blockIdx.x * blockDim.x + threadIdx.x;
    if (i >= 128 * 128) return;
    int r = i >> 7, c = i & 127;
    int hr = r >> 5, hc = c >> 5;
    W[i] = (hr == hc) ? rel[(hr << 10) + ((r & 31) << 5) + (c & 31)] : 0.0f;
}

__global__ void fill_kernel(float* __restrict__ p, float v, int n)
{
    int i = blockIdx.x * blockDim.x + threadIdx.x;
    if (i < n) p[i] = v;
}

// one thread per (edge, head): alpha = <q[dst], k_rel[src]> * p_rel * 1/sqrt(D)
__global__ void edge_alpha_kernel(const float* __restrict__ q,
                                  const float* __restrict__ krel,
                                  const int* __restrict__ src,
                                  const int* __restrict__ dst,
                                  const float* __restrict__ p_rel,
                                  float* __restrict__ alpha,
                                  float* segmax, int E)
{
    int idx = blockIdx.x * blockDim.x + threadIdx.x;
    if (idx >= E * NHEADS) return;
    int e = idx >> 2, h = idx & 3;
    int s = src[e], d = dst[e];
    const float4* qr = (const float4*)(q    + (size_t)d * 128 + h * 32);
    const float4* kr = (const float4*)(krel + (size_t)s * 128 + h * 32);
    float sum = 0.0f;
    #pragma unroll
    for (int i = 0; i < 8; i++) {
        float4 a = qr[i], b = kr[i];
        sum += a.x * b.x + a.y * b.y + a.z * b.z + a.w * b.w;
    }
    float aval = sum * p_rel[h] * 0.17677669529663687f;  // 1/sqrt(32)
    alpha[idx] = aval;
    atomicMaxF(&segmax[(size_t)d * 4 + h], aval);
}

// one thread per (edge, head): w = exp(a - m); sum += w; agg[dst] += w * v_rel[src]
__global__ void edge_agg_kernel(const float* __restrict__ alpha,
                                const float* __restrict__ vrel,
                                const int* __restrict__ src,
                                const int* __restrict__ dst,
                                const float* __restrict__ segmax,
                                float* segsum, float* agg, int E)
{
    int idx = blockIdx.x * blockDim.x + threadIdx.x;
    if (idx >= E * NHEADS) return;
    int e = idx >> 2, h = idx & 3;
    int s = src[e], d = dst[e];
    float m = segmax[(size_t)d * 4 + h];
    float w = expf(alpha[idx] - m);
    atomicAdd(&segsum[(size_t)d * 4 + h], w);
    const float4* vr = (const float4*)(vrel + (size_t)s * 128 + h * 32);
    float* ag = agg + (size_t)d * 128 + h * 32;
    #pragma unroll
    for (int i = 0; i < 8; i++) {
        float4 v = vr[i];
        atomicAdd(ag + i * 4 + 0, w * v.x);
        atomicAdd(ag + i * 4 + 1, w * v.y);
        atomicAdd(ag + i * 4 + 2, w * v.z);
        atomicAdd(ag + i * 4 + 3, w * v.w);
    }
}

// agg = gelu( agg / (sum + 1e-16) ), exact erf form
__global__ void norm_gelu_kernel(float* __restrict__ agg,
                                 const float* __restrict__ segsum, int total)
{
    int i = blockIdx.x * blockDim.x + threadIdx.x;
    if (i >= total) return;
    int node = i >> 7;
    int h = (i >> 5) & 3;
    float s = segsum[node * 4 + h];
    float x = agg[i] / (s + 1e-16f);
    agg[i] = 0.5f * x * (1.0f + erff(x * 0.70710678118654752f));
}

extern "C" void kernel_launch(void* const* d_in, const int* in_sizes, int n_in,
                              void* d_out, int out_size, void* d_ws, size_t ws_size,
                              hipStream_t stream)
{
    (void)in_sizes; (void)out_size; (void)ws_size;
    auto F = [&](int i) { return (const float*)d_in[i]; };

    // setup_inputs() dict insertion order
    const float* x_author = F(0);
    const float* x_paper  = F(1);
    const float* lin_a_w = F(2); const float* lin_a_b = F(3);
    const float* lin_p_w = F(4); const float* lin_p_b = F(5);
    const float* out_w   = F(6); const float* out_b   = F(7);
    const int* ei_w  = (const int*)d_in[n_in - 3];   // writes: author->paper
    const int* ei_wb = (const int*)d_in[n_in - 2];   // written_by: paper->author
    const int* ei_c  = (const int*)d_in[n_in - 1];   // cites: paper->paper

    // ---- workspace carve (floats) ----
    float* ws = (float*)d_ws;
    size_t off = 0;
    auto alloc = [&](size_t n) { float* p = ws + off; off += n; return p; };
    float* xa  = alloc((size_t)RA * FDIM); float* xp  = alloc((size_t)RP * FDIM);
    float* xa2 = alloc((size_t)RA * FDIM); float* xp2 = alloc((size_t)RP * FDIM);
    float* ka  = alloc((size_t)RA * FDIM); float* qa  = alloc((size_t)RA * FDIM);
    float* va  = alloc((size_t)RA * FDIM);
    float* kp  = alloc((size_t)RP * FDIM); float* qp  = alloc((size_t)RP * FDIM);
    float* vp  = alloc((size_t)RP * FDIM);
    float* krel_wb = alloc((size_t)RP * FDIM);
    float* vrel_wb = alloc((size_t)RP * FDIM);
    float* Wtmp    = alloc((size_t)FDIM * FDIM);
    float* alpha_p = alloc((size_t)2 * EDG * NHEADS);
    float* alpha_a = alloc((size_t)EDG * NHEADS);
    float* segmax_p = alloc((size_t)RP * NHEADS); float* segsum_p = alloc((size_t)RP * NHEADS);
    float* segmax_a = alloc((size_t)RA * NHEADS); float* segsum_a = alloc((size_t)RA * NHEADS);
    float* agg_a = alloc((size_t)RA * FDIM);
    float* agg_p = alloc((size_t)RP * FDIM);

    const int GA = (RA + 127) / 128;
    const int GP = (RP + 127) / 128;
    const int EGRID  = (EDG * NHEADS + 255) / 256;
    const int EXGRID = (FDIM * FDIM + 255) / 256;
    const float NEG_INF = -__builtin_inff();

#define GEMM(Nv, MODEv, X_, W_, B_, RES_, GATE_, Y_, M_, GRID_) \
    gemm128_kernel<Nv, MODEv><<<GRID_, 256, 0, stream>>>(X_, W_, B_, RES_, GATE_, Y_, M_)

    // input linear + relu
    GEMM(128, 1, x_author, lin_a_w, lin_a_b, nullptr, nullptr, xa, RA, GA);
    GEMM(128, 1, x_paper,  lin_p_w, lin_p_b, nullptr, nullptr, xp, RP, GP);

    for (int l = 0; l < 2; l++) {
        const int pb = 8 + l * 27;
        const float *kaw = F(pb+0),  *kab = F(pb+1),  *kpw = F(pb+2),  *kpb = F(pb+3);
        const float *qaw = F(pb+4),  *qab = F(pb+5),  *qpw = F(pb+6),  *qpb = F(pb+7);
        const float *vaw = F(pb+8),  *vab = F(pb+9),  *vpw = F(pb+10), *vpb = F(pb+11);
        const float *aaw = F(pb+12), *aab = F(pb+13), *apw = F(pb+14), *apb = F(pb+15);
        const float *skipA = F(pb+16), *skipP = F(pb+17);
        const float *arel_w  = F(pb+18), *mrel_w  = F(pb+19), *prel_w  = F(pb+20);
        const float *arel_wb = F(pb+21), *mrel_wb = F(pb+22), *prel_wb = F(pb+23);
        const float *arel_c  = F(pb+24), *mrel_c  = F(pb+25), *prel_c  = F(pb+26);

        // K/Q/V projections
        GEMM(128, 0, xa, kaw, kab, nullptr, nullptr, ka, RA, GA);
        GEMM(128, 0, xa, qaw, qab, nullptr, nullptr, qa, RA, GA);
        GEMM(128, 0, xa, vaw, vab, nullptr, nullptr, va, RA, GA);
        GEMM(128, 0, xp, kpw, kpb, nullptr, nullptr, kp, RP, GP);
        GEMM(128, 0, xp, qpw, qpb, nullptr, nullptr, qp, RP, GP);
        GEMM(128, 0, xp, vpw, vpb, nullptr, nullptr, vp, RP, GP);

        // relation transforms as block-diag GEMMs (author in-place; paper: wb to
        // buffers first, then cites in-place)
        expand_blockdiag_kernel<<<EXGRID, 256, 0, stream>>>(arel_w, Wtmp);
        GEMM(128, 0, ka, Wtmp, nullptr, nullptr, nullptr, ka, RA, GA);      // k_rel writes
        expand_blockdiag_kernel<<<EXGRID, 256, 0, stream>>>(mrel_w, Wtmp);
        GEMM(128, 0, va, Wtmp, nullptr, nullptr, nullptr, va, RA, GA);      // v_rel writes
        expand_blockdiag_kernel<<<EXGRID, 256, 0, stream>>>(arel_wb, Wtmp);
        GEMM(128, 0, kp, Wtmp, nullptr, nullptr, nullptr, krel_wb, RP, GP); // k_rel written_by
        expand_blockdiag_kernel<<<EXGRID, 256, 0, stream>>>(mrel_wb, Wtmp);
        GEMM(128, 0, vp, Wtmp, nullptr, nullptr, nullptr, vrel_wb, RP, GP); // v_rel written_by
        expand_blockdiag_kernel<<<EXGRID, 256, 0, stream>>>(arel_c, Wtmp);
        GEMM(128, 0, kp, Wtmp, nullptr, nullptr, nullptr, kp, RP, GP);      // k_rel cites
        expand_blockdiag_kernel<<<EXGRID, 256, 0, stream>>>(mrel_c, Wtmp);
        GEMM(128, 0, vp, Wtmp, nullptr, nullptr, nullptr, vp, RP, GP);      // v_rel cites

        // reset segment state (every call: graph-replay safe)
        fill_kernel<<<(RP*NHEADS+255)/256, 256, 0, stream>>>(segmax_p, NEG_INF, RP*NHEADS);
        fill_kernel<<<(RP*NHEADS+255)/256, 256, 0, stream>>>(segsum_p, 0.0f,   RP*NHEADS);
        fill_kernel<<<(RA*NHEADS+255)/256, 256, 0, stream>>>(segmax_a, NEG_INF, RA*NHEADS);
        fill_kernel<<<(RA*NHEADS+255)/256, 256, 0, stream>>>(segsum_a, 0.0f,   RA*NHEADS);
        fill_kernel<<<(RP*FDIM+255)/256, 256, 0, stream>>>(agg_p, 0.0f, RP*FDIM);
        fill_kernel<<<(RA*FDIM+255)/256, 256, 0, stream>>>(agg_a, 0.0f, RA*FDIM);

        // attention logits + per-destination max (softmax shared across edge types)
        edge_alpha_kernel<<<EGRID, 256, 0, stream>>>(qp, ka,      ei_w,  ei_w + EDG,  prel_w,  alpha_p,                       segmax_p, EDG);
        edge_alpha_kernel<<<EGRID, 256, 0, stream>>>(qp, kp,      ei_c,  ei_c + EDG,  prel_c,  alpha_p + (size_t)EDG*NHEADS, segmax_p, EDG);
        edge_alpha_kernel<<<EGRID, 256, 0, stream>>>(qa, krel_wb, ei_wb, ei_wb + EDG, prel_wb, alpha_a,                       segmax_a, EDG);

        // exp weights, segment sums, weighted message aggregation
        edge_agg_kernel<<<EGRID, 256, 0, stream>>>(alpha_p,                       va,      ei_w,  ei_w + EDG,  segmax_p, segsum_p, agg_p, EDG);
        edge_agg_kernel<<<EGRID, 256, 0, stream>>>(alpha_p + (size_t)EDG*NHEADS, vp,      ei_c,  ei_c + EDG,  segmax_p, segsum_p, agg_p, EDG);
        edge_agg_kernel<<<EGRID, 256, 0, stream>>>(alpha_a,                       vrel_wb, ei_wb, ei_wb + EDG, segmax_a, segsum_a, agg_a, EDG);

        // normalize + exact gelu
        norm_gelu_kernel<<<(RA*FDIM+255)/256, 256, 0, stream>>>(agg_a, segsum_a, RA*FDIM);
        norm_gelu_kernel<<<(RP*FDIM+255)/256, 256, 0, stream>>>(agg_p, segsum_p, RP*FDIM);

        // 'a' linear + sigmoid-gated skip connection
        GEMM(128, 2, agg_a, aaw, aab, xa, skipA, xa2, RA, GA);
        GEMM(128, 2, agg_p, apw, apb, xp, skipP, xp2, RP, GP);

        { float* t = xa; xa = xa2; xa2 = t; }
        { float* t = xp; xp = xp2; xp2 = t; }
    }

    // output linear on author nodes -> d_out [20000,64]
    GEMM(64, 0, xa, out_w, out_b, nullptr, nullptr, (float*)d_out, RA, GA);
#undef GEMM
}